// CenterNet_11982958756181
// MI455X (gfx1250) — compile-verified
//
#include <hip/hip_runtime.h>
#include <stdint.h>

// CenterNet decode for MI455X (gfx1250, wave32).
// HBM floor: ~176 MB / 23.3 TB/s ~= 7.6 us. No matmul content -> no WMMA;
// the CDNA5 features that matter are the async data movers:
//  - kernel 1: Tensor Data Mover (tensor_load_to_lds + TENSORcnt) streams the
//    64KB class plane into LDS with ONE instruction per workgroup
//    (fallback: per-lane global_load_async_to_lds_b128 + ASYNCcnt)
//  - kernel 2: per-lane async global->LDS b128 copies (ASYNCcnt)
// All NMS + top-K selection then runs on-chip (LDS + registers).

#define Bn 32
#define Cn 80
#define Hn 128
#define Wn 128
#define HWn (Hn * Wn)
#define Kn 100
#define CKn (Cn * Kn)

#ifndef __has_builtin
#define __has_builtin(x) 0
#endif

#if __has_builtin(__builtin_amdgcn_tensor_load_to_lds)
#define HAVE_TDM 1
#else
#define HAVE_TDM 0
#endif

typedef unsigned int u32x4 __attribute__((ext_vector_type(4)));
typedef int i32x4 __attribute__((ext_vector_type(4)));
typedef int i32x8 __attribute__((ext_vector_type(8)));

// ---- CDNA5 async global->LDS copy (per-lane b128), tracked by ASYNCcnt ----
__device__ __forceinline__ void async_b128(uint32_t lds_addr, uint32_t voff,
                                           const void* sbase) {
  asm volatile("global_load_async_to_lds_b128 %0, %1, %2"
               :: "v"(lds_addr), "v"(voff), "s"(sbase) : "memory");
}
__device__ __forceinline__ void wait_async0() {
  asm volatile("s_wait_asynccnt 0" ::: "memory");
}
// Low 32 bits of a generic pointer to LDS == LDS byte offset (ISA 10.2 aperture).
__device__ __forceinline__ uint32_t lds_off32(const void* p) {
  return (uint32_t)(uintptr_t)p;
}

#if HAVE_TDM
// ---- Tensor Data Mover: 1-row tile of `ndw` dwords, global -> LDS ----
// D# per ISA ch.8: group0 = {count=1 | lds_addr | global_addr(57b) | type=2},
// group1 = {mask=0, data_size=4B, dim0=tile0=ndw, dim1=1, stride=ndw},
// groups 2/3 zero (<=2D tensor, no gather/iterate/pad/barrier).
__device__ __forceinline__ void tdm_load_1d_issue(uint32_t lds_addr,
                                                  const void* gptr,
                                                  uint32_t ndw) {
  uint64_t ga = (uint64_t)(uintptr_t)gptr;
  u32x4 g0 = {1u,                                   // count=1 (valid), user
              lds_addr,                             // lds_addr [63:32]
              (uint32_t)ga,                         // global_addr[31:0]
              ((uint32_t)(ga >> 32) & 0x01FFFFFFu)  // global_addr[56:32]
                  | (2u << 30)};                    // type=2 ("image")
  i32x8 g1 = {(int)(2u << 16),                      // data_size=2 (4 bytes)
              (int)((ndw & 0xFFFFu) << 16),         // tensor_dim0[15:0]
              (int)((ndw >> 16) | (1u << 16)),      // dim0[31:16] | dim1.lo=1
              (int)((ndw & 0xFFFFu) << 16),         // dim1.hi=0 | tile_dim0
              0,                                    // tile_dim1=0, tile_dim2=0
              (int)ndw,                             // tensor_dim0_stride[31:0]
              0, 0};
  i32x4 gz = {0, 0, 0, 0};
#if __clang_major__ >= 23
  i32x8 gz8 = {0, 0, 0, 0, 0, 0, 0, 0};
  __builtin_amdgcn_tensor_load_to_lds(g0, g1, gz, gz, gz8, 0);
#else
  __builtin_amdgcn_tensor_load_to_lds(g0, g1, gz, gz, 0);
#endif
}
#endif

// ---- block-wide u32 sum over 256 threads (LDS tree) ----
__device__ __forceinline__ uint32_t block_sum256(uint32_t v, uint32_t* red, int tid) {
  red[tid] = v;
  __syncthreads();
#pragma unroll
  for (int s = 128; s > 0; s >>= 1) {
    if (tid < s) red[tid] += red[tid + s];
    __syncthreads();
  }
  uint32_t t = red[0];
  __syncthreads();
  return t;
}

// ---- 256-element bitonic sort, descending, one element per thread ----
__device__ __forceinline__ void bitonic256_desc(unsigned long long* cand, int tid) {
  for (int k = 2; k <= 256; k <<= 1) {
    for (int j = k >> 1; j > 0; j >>= 1) {
      int ixj = tid ^ j;
      if (ixj > tid) {
        unsigned long long a = cand[tid], b = cand[ixj];
        if (((tid & k) == 0) ? (a < b) : (a > b)) {
          cand[tid] = b;
          cand[ixj] = a;
        }
      }
      __syncthreads();
    }
  }
}

// =======================================================================
// Kernel 1: per-(b,c) 3x3 NMS + top-100.  grid = B*C, block = 256.
// =======================================================================
__global__ __launch_bounds__(256)
void k1_nms_topk(const float* __restrict__ fmap,
                 float* __restrict__ scores_ws,
                 int* __restrict__ inds_ws) {
  __shared__ float plane[HWn];                 // 64 KB: one class plane
  __shared__ unsigned long long surv[2048];    // 16 KB: NMS survivors (keys)
  __shared__ unsigned long long cand[256];     //  2 KB: threshold candidates
  __shared__ uint32_t red[256];                //  1 KB: reduction scratch
  __shared__ uint32_t scnt, ccnt;

  const int tid = threadIdx.x;
  const int bc = blockIdx.x;                   // b*Cn + c
  const float* src = fmap + (size_t)bc * HWn;
  uint32_t lbase = lds_off32((const void*)plane);

#if HAVE_TDM
  // One TDM descriptor streams the whole 64KB plane into LDS.
  if (tid < 32) tdm_load_1d_issue(lbase, (const void*)src, (uint32_t)HWn);
#else
  // Fallback: 4096 per-lane async b128 transfers.
#pragma unroll
  for (int i = 0; i < 16; ++i) {
    uint32_t off = (uint32_t)((i * 256 + tid) * 16);
    async_b128(lbase + off, off, (const void*)src);
  }
#endif

#pragma unroll
  for (int i = 0; i < 8; ++i) surv[tid + i * 256] = 0ull;
  if (tid == 0) { scnt = 0u; ccnt = 0u; }

#if HAVE_TDM
  if (tid < 32) __builtin_amdgcn_s_wait_tensorcnt(0);
#else
  wait_async0();
#endif
  __syncthreads();

  // 3x3 NMS (keep iff v == 3x3 max, -inf padding == bounds check).
  // Survivors appended as (value_bits<<32)|~index: monotone for positive
  // floats, ties resolve to lowest spatial index like jax.lax.top_k.
  for (int j = 0; j < 64; ++j) {
    int e = tid + j * 256;
    int x = e & (Wn - 1), y = e >> 7;
    float v = plane[e];
    float m = v;
#pragma unroll
    for (int dy = -1; dy <= 1; ++dy) {
#pragma unroll
      for (int dx = -1; dx <= 1; ++dx) {
        int yy = y + dy, xx = x + dx;
        if (yy >= 0 && yy < Hn && xx >= 0 && xx < Wn)
          m = fmaxf(m, plane[yy * Wn + xx]);
      }
    }
    if (v == m) {
      uint32_t p = atomicAdd(&scnt, 1u);
      if (p < 2048u)
        surv[p] = ((unsigned long long)__float_as_uint(v) << 32) |
                  (uint32_t)(~e);
    }
  }
  __syncthreads();

  // Survivor keys -> registers (8 per thread).
  unsigned long long ks[8];
#pragma unroll
  for (int j = 0; j < 8; ++j) ks[j] = surv[tid + j * 256];

  // Binary search the 100th-largest value-bit threshold T.
  uint32_t lo = 0u, hi = 0x3F800001u;          // scores in [0,1)
  while (hi - lo > 1u) {
    uint32_t mid = lo + ((hi - lo) >> 1);
    uint32_t c = 0;
#pragma unroll
    for (int j = 0; j < 8; ++j)
      c += ((uint32_t)(ks[j] >> 32) >= mid) ? 1u : 0u;
    uint32_t tot = block_sum256(c, red, tid);
    if (tot >= (uint32_t)Kn) lo = mid; else hi = mid;
  }
  const uint32_t T = lo;

  // Collect >= T (>=100, <100+ties), sort descending, emit top-100.
  cand[tid] = 0ull;
  __syncthreads();
#pragma unroll
  for (int j = 0; j < 8; ++j) {
    unsigned long long key = ks[j];
    if (key != 0ull && (uint32_t)(key >> 32) >= T) {
      uint32_t p = atomicAdd(&ccnt, 1u);
      if (p < 256u) cand[p] = key;
    }
  }
  __syncthreads();
  bitonic256_desc(cand, tid);

  if (tid < Kn) {
    unsigned long long key = cand[tid];
    scores_ws[(size_t)bc * Kn + tid] = __uint_as_float((uint32_t)(key >> 32));
    inds_ws[(size_t)bc * Kn + tid] = (int)(~(uint32_t)key) & (HWn - 1);
  }
}

// =======================================================================
// Kernel 2: per-batch global top-100 over C*K=8000 + bbox decode.
// grid = B, block = 256.
// =======================================================================
__global__ __launch_bounds__(256)
void k2_global_topk(const float* __restrict__ scores_ws,
                    const int* __restrict__ inds_ws,
                    const float* __restrict__ wh,
                    const float* __restrict__ regp,
                    float* __restrict__ out) {
  __shared__ float vals[CKn];                  // 32000 B
  __shared__ unsigned long long cand[256];
  __shared__ uint32_t red[256];
  __shared__ uint32_t ccnt;

  const int tid = threadIdx.x;
  const int b = blockIdx.x;
  const float* src = scores_ws + (size_t)b * CKn;

  // Per-lane async global->LDS copies (ASYNCcnt path).
  uint32_t lbase = lds_off32((const void*)vals);
  for (int i = tid; i < (CKn * 4) / 16; i += 256) {   // 2000 b128 transfers
    uint32_t off = (uint32_t)(i * 16);
    async_b128(lbase + off, off, (const void*)src);
  }
  if (tid == 0) ccnt = 0u;
  wait_async0();
  __syncthreads();

  // Pack keys into registers: (bits<<32)|~pos, pos in [0, 8000).
  unsigned long long ks[32];
#pragma unroll
  for (int j = 0; j < 32; ++j) {
    int pos = tid + j * 256;
    if (pos < CKn) {
      float v = vals[pos];
      ks[j] = ((unsigned long long)__float_as_uint(v) << 32) |
              (uint32_t)(~pos);
    } else {
      ks[j] = 0ull;
    }
  }

  uint32_t lo = 0u, hi = 0x3F800001u;
  while (hi - lo > 1u) {
    uint32_t mid = lo + ((hi - lo) >> 1);
    uint32_t c = 0;
#pragma unroll
    for (int j = 0; j < 32; ++j)
      c += ((uint32_t)(ks[j] >> 32) >= mid) ? 1u : 0u;
    uint32_t tot = block_sum256(c, red, tid);
    if (tot >= (uint32_t)Kn) lo = mid; else hi = mid;
  }
  const uint32_t T = lo;

  cand[tid] = 0ull;
  __syncthreads();
#pragma unroll
  for (int j = 0; j < 32; ++j) {
    unsigned long long key = ks[j];
    if (key != 0ull && (uint32_t)(key >> 32) >= T) {
      uint32_t p = atomicAdd(&ccnt, 1u);
      if (p < 256u) cand[p] = key;
    }
  }
  __syncthreads();
  bitonic256_desc(cand, tid);

  // Decode + write: bboxes [B,K,4] | scores [B,K,1] | clses [B,K,1].
  if (tid < Kn) {
    unsigned long long key = cand[tid];
    float score = __uint_as_float((uint32_t)(key >> 32));
    uint32_t pos = ~(uint32_t)key;
    if (pos >= (uint32_t)CKn) pos = 0u;
    int cls = (int)(pos / (uint32_t)Kn);       // idx // K
    int ind = inds_ws[(size_t)b * CKn + pos] & (HWn - 1);

    float xs = (float)(ind & (Wn - 1));        // ind % W
    float ys = (float)(ind >> 7);              // ind / W
    const float* regb = regp + (size_t)b * 2 * HWn;
    const float* whb = wh + (size_t)b * 2 * HWn;
    float xc = xs + regb[ind];                 // + reg[b,0,ind]
    float yc = ys + regb[HWn + ind];           // + reg[b,1,ind]
    float hw = whb[ind] * 0.5f;
    float hh = whb[HWn + ind] * 0.5f;

    float* obb = out + (size_t)(b * Kn + tid) * 4;
    obb[0] = xc - hw;
    obb[1] = yc - hh;
    obb[2] = xc + hw;
    obb[3] = yc + hh;
    out[(size_t)Bn * Kn * 4 + (size_t)b * Kn + tid] = score;
    out[(size_t)Bn * Kn * 4 + (size_t)Bn * Kn + (size_t)b * Kn + tid] =
        (float)cls;
  }
}

extern "C" void kernel_launch(void* const* d_in, const int* in_sizes, int n_in,
                              void* d_out, int out_size, void* d_ws,
                              size_t ws_size, hipStream_t stream) {
  (void)in_sizes; (void)n_in; (void)out_size; (void)ws_size;
  const float* fmap = (const float*)d_in[0];
  const float* wh = (const float*)d_in[1];
  const float* regp = (const float*)d_in[2];
  // d_in[3] is K (=100), baked in at compile time (output shape fixed).

  float* scores_ws = (float*)d_ws;                                   // B*C*K f32
  int* inds_ws = (int*)((char*)d_ws +
                        (size_t)Bn * Cn * Kn * sizeof(float));       // B*C*K i32

  k1_nms_topk<<<Bn * Cn, 256, 0, stream>>>(fmap, scores_ws, inds_ws);
  k2_global_topk<<<Bn, 256, 0, stream>>>(scores_ws, inds_ws, wh, regp,
                                         (float*)d_out);
}